// NPIDLoss_11287174054161
// MI455X (gfx1250) — compile-verified
//
#include <hip/hip_runtime.h>

typedef float v2f __attribute__((ext_vector_type(2)));
typedef float v8f __attribute__((ext_vector_type(8)));

#define BATCH    256
#define KTOT     4097            // NEGS + 1
#define DIM      128
#define NTOT     1281167
#define TILES    257             // ceil(KTOT / 16)
#define WAVES_TOTAL (BATCH * TILES)        // 65792
#define BLOCKS1  (WAVES_TOTAL / 4)         // 16448 blocks of 128 threads (4 waves)
#define NELEM    (BATCH * KTOT)            // 1048832
#define BLOCKS3  1024
#define TEMP_INV 14.2857142857142857f      // 1 / 0.07
#define M_CONST  (4096.0f / 1281167.0f)

// ---------------------------------------------------------------------------
// Kernel 1: gather 16 bank rows per wave, compute sim via chained f32 WMMA,
// write exp(sim/T), accumulate per-block sum of exp.
// Wave w -> (b = w / TILES, tile t = w % TILES), k0 = 16*t.
// WMMA 16x16x4 f32: A 16x4 (2 VGPR), B 4x16 (2 VGPR), C/D 16x16 (8 VGPR).
// B columns all hold proj[b] chunk -> every column of D equals sim.
// Dim->K mapping chosen so each lane feeds 2 WMMAs from one b128 load:
//   lane<16 : float4 dims [8j .. 8j+3]   -> K0,K1 slots of steps 2j, 2j+1
//   lane>=16: float4 dims [8j+4 .. 8j+7] -> K2,K3 slots of steps 2j, 2j+1
// ---------------------------------------------------------------------------
__global__ __launch_bounds__(128) void npid_sim(
    const float* __restrict__ proj, const float* __restrict__ bank,
    const int* __restrict__ idx, float* __restrict__ exp_buf,
    float* __restrict__ partials) {
  __shared__ float wsum[4];
  const int lane = threadIdx.x & 31;
  const int warp = threadIdx.x >> 5;
  const int gw   = blockIdx.x * 4 + warp;
  const int b    = gw / TILES;
  const int t    = gw - b * TILES;
  const int k0   = t * 16;
  const int li   = lane & 15;       // row within tile (A: M = li)
  const int hi   = lane >> 4;       // half: K-slot group

  int k  = k0 + li;
  int kc = (k < KTOT) ? k : k0;     // k0 <= 4096 is always valid
  const long long row = (long long)idx[b * KTOT + kc];
  const float* __restrict__ arow = bank + row * DIM + (hi << 2);
  const float* __restrict__ prow = proj + b * DIM + (hi << 2);

  v8f c = {0.f, 0.f, 0.f, 0.f, 0.f, 0.f, 0.f, 0.f};
#pragma unroll
  for (int j = 0; j < 16; ++j) {
    const float4 a4 = *(const float4*)(arow + j * 8);
    const float4 p4 = *(const float4*)(prow + j * 8);
    v2f a0 = {a4.x, a4.y}, a1 = {a4.z, a4.w};
    v2f b0 = {p4.x, p4.y}, b1 = {p4.z, p4.w};
    c = __builtin_amdgcn_wmma_f32_16x16x4_f32(false, a0, false, b0,
                                              (short)0, c, false, false);
    c = __builtin_amdgcn_wmma_f32_16x16x4_f32(false, a1, false, b1,
                                              (short)0, c, false, false);
  }

  // D layout: lanes 0-15 hold rows 0-7 in c[0..7]; lanes 16-31 hold rows 8-15.
  // All 16 columns identical. Lane with li<8 extracts row hi*8 + li.
  float myexp = 0.f;
  if (li < 8) {
    float sv = c[0];
#pragma unroll
    for (int v = 1; v < 8; ++v) sv = (li == v) ? c[v] : sv;
    const int myk = k0 + hi * 8 + li;
    if (myk < KTOT) {
      const float e = expf(sv * TEMP_INV);
      exp_buf[b * KTOT + myk] = e;
      myexp = e;
    }
  }
  // deterministic wave reduction
#pragma unroll
  for (int off = 16; off; off >>= 1) myexp += __shfl_xor(myexp, off, 32);
  if (lane == 0) wsum[warp] = myexp;
  __syncthreads();
  if (threadIdx.x == 0)
    partials[blockIdx.x] = (wsum[0] + wsum[1]) + (wsum[2] + wsum[3]);
}

// ---------------------------------------------------------------------------
// Kernel 2: fixed-order reduce of block partials -> z = mean(exp) * N
// ---------------------------------------------------------------------------
__global__ __launch_bounds__(256) void npid_reduce_z(
    const float* __restrict__ partials, int n, float* __restrict__ zout) {
  __shared__ float sm[256];
  float s = 0.f;
  for (int i = threadIdx.x; i < n; i += 256) s += partials[i];
  sm[threadIdx.x] = s;
  __syncthreads();
  for (int st = 128; st; st >>= 1) {
    if ((int)threadIdx.x < st) sm[threadIdx.x] += sm[threadIdx.x + st];
    __syncthreads();
  }
  if (threadIdx.x == 0)
    zout[0] = (sm[0] / (float)NELEM) * (float)NTOT;
}

// ---------------------------------------------------------------------------
// Kernel 3: per-element log terms, per-block partial sums.
// k==0: log(o/(o+m)) ; k>0: log(m/(o+m)) with o = exp_buf/z
// ---------------------------------------------------------------------------
__global__ __launch_bounds__(256) void npid_terms(
    const float* __restrict__ exp_buf, const float* __restrict__ zp,
    float* __restrict__ partials2) {
  __shared__ float sm[256];
  const float z = zp[0];
  float s = 0.f;
  for (int i = blockIdx.x * 256 + threadIdx.x; i < NELEM; i += 256 * BLOCKS3) {
    const int bb = i / KTOT;
    const int k  = i - bb * KTOT;
    const float o = exp_buf[i] / z;
    const float num = (k == 0) ? o : M_CONST;
    s += logf(num / (o + M_CONST));
  }
  sm[threadIdx.x] = s;
  __syncthreads();
  for (int st = 128; st; st >>= 1) {
    if ((int)threadIdx.x < st) sm[threadIdx.x] += sm[threadIdx.x + st];
    __syncthreads();
  }
  if (threadIdx.x == 0) partials2[blockIdx.x] = sm[0];
}

// ---------------------------------------------------------------------------
// Kernel 4: final reduce -> loss = -sum / BATCH
// ---------------------------------------------------------------------------
__global__ __launch_bounds__(256) void npid_finalize(
    const float* __restrict__ partials2, float* __restrict__ out) {
  __shared__ float sm[256];
  const int t = threadIdx.x;
  float s = (partials2[t] + partials2[t + 256]) +
            (partials2[t + 512] + partials2[t + 768]);
  sm[t] = s;
  __syncthreads();
  for (int st = 128; st; st >>= 1) {
    if (t < st) sm[t] += sm[t + st];
    __syncthreads();
  }
  if (t == 0) out[0] = -sm[0] / (float)BATCH;
}

// ---------------------------------------------------------------------------
// Kernel 5: pos_samples = bank[idx[:,0]]  -> d_out[1 ..]
// ---------------------------------------------------------------------------
__global__ __launch_bounds__(128) void npid_poscopy(
    const float* __restrict__ bank, const int* __restrict__ idx,
    float* __restrict__ out_pos) {
  const int b = blockIdx.x;
  const int d = threadIdx.x;
  const long long r = (long long)idx[b * KTOT];
  out_pos[b * DIM + d] = bank[r * DIM + d];
}

extern "C" void kernel_launch(void* const* d_in, const int* in_sizes, int n_in,
                              void* d_out, int out_size, void* d_ws,
                              size_t ws_size, hipStream_t stream) {
  (void)in_sizes; (void)n_in; (void)out_size; (void)ws_size;
  const float* proj = (const float*)d_in[0];   // (256, 128) f32
  const float* bank = (const float*)d_in[1];   // (1281167, 128) f32
  const int*   idx  = (const int*)d_in[2];     // (256, 4097) i32
  float* out = (float*)d_out;                  // [loss, pos_samples(256x128)]

  float* ws        = (float*)d_ws;
  float* exp_buf   = ws;                       // NELEM floats
  float* partials  = exp_buf + NELEM;          // BLOCKS1 floats
  float* partials2 = partials + BLOCKS1;       // BLOCKS3 floats
  float* zscal     = partials2 + BLOCKS3;      // 1 float

  npid_sim<<<BLOCKS1, 128, 0, stream>>>(proj, bank, idx, exp_buf, partials);
  npid_reduce_z<<<1, 256, 0, stream>>>(partials, BLOCKS1, zscal);
  npid_terms<<<BLOCKS3, 256, 0, stream>>>(exp_buf, zscal, partials2);
  npid_finalize<<<1, 256, 0, stream>>>(partials2, out);
  npid_poscopy<<<BATCH, 128, 0, stream>>>(bank, idx, out + 1);
}